// AttnDecoderRNN_19954418057554
// MI455X (gfx1250) — compile-verified
//
#include <hip/hip_runtime.h>
#include <math.h>

// ---------------------------------------------------------------------------
// AttnDecoderRNN single step on gfx1250 (CDNA5, wave32).
// All GEMVs run through V_WMMA_F32_16X16X4_F32: 16 weight rows per wave as the
// A matrix, activation chunk broadcast across B columns, column 0 of D = the
// 16 dot products. Memory-bound problem (~241 MB / call, ~0.5 FLOP/byte), so
// the design goal is: stream each weight matrix once at full bandwidth, keep
// the hot 150 MB of out_w resident in the 192 MB L2 across graph replays by
// marking the overflow tail non-temporal.
// ---------------------------------------------------------------------------

#define HID 1024
#define VOC 50257
#define LEN 100

// out_w rows [0, NT_SPLIT) use RT loads (stay L2-resident across replays,
// ~151 MB); rows [NT_SPLIT, VOC) use non-temporal loads so the streaming tail
// cannot evict the resident portion. ~35 MB of L2 is left for the other
// weight matrices, activations and logits.
#define NT_SPLIT 36864

typedef __attribute__((ext_vector_type(2))) float v2f;
typedef __attribute__((ext_vector_type(8))) float v8f;

// workspace layout (float offsets)
#define WS_XIN   0        // [emb | h0]            : 2048
#define WS_XCAT  2048     // [emb | attn_applied]  : 2048
#define WS_X     4096     // combined+relu         : 1024
#define WS_GI    5120     // x @ w_ih^T + b_ih     : 3072
#define WS_GH    8192     // h0 @ w_hh^T + b_hh    : 3072
#define WS_HNEW  11264    // new hidden            : 1024
#define WS_LOGZ  12288    // scalar

// Wave-cooperative GEMV: out[row0+m] = dot(W[row0+m,0:K], xs[0:K]) + bias[row0+m].
// A-matrix layout (16x4 f32): lanes 0-15 carry K={0,1}, lanes 16-31 K={2,3};
// D column 0 lands in lane 0 (rows 0..7, acc[0..7]) and lane 16 (rows 8..15).
template <bool NT>
__device__ __forceinline__ void gemv16_wave(const float* __restrict__ W,
                                            const float* __restrict__ xs,
                                            const float* __restrict__ bias,
                                            float* __restrict__ out,
                                            int row0, int nrows, int K, int relu) {
    const int lane  = threadIdx.x & 31;
    const int half  = lane >> 4;           // 0 or 1
    const int khalf = half << 1;           // 0 or 2
    int m = row0 + (lane & 15);
    if (m > nrows - 1) m = nrows - 1;      // clamp loads for ragged last group
    const float* wrow = W + (size_t)m * (size_t)K + khalf;

    v8f acc = {};
#pragma unroll 16
    for (int k0 = 0; k0 < K; k0 += 4) {
        v2f a = NT ? __builtin_nontemporal_load((const v2f*)(wrow + k0))
                   : *(const v2f*)(wrow + k0);
        v2f b = *(const v2f*)(xs + k0 + khalf);
        acc = __builtin_amdgcn_wmma_f32_16x16x4_f32(
            /*neg_a=*/false, a, /*neg_b=*/false, b,
            /*c_mod=*/(short)0, acc, /*reuse_a=*/false, /*reuse_b=*/false);
    }

    if ((lane & 15) == 0) {                // lane 0 -> rows 0..7, lane 16 -> 8..15
        const int base = row0 + half * 8;
        if (row0 + 16 <= nrows) {          // full group: unconditional vector path
            float t[8];
#pragma unroll
            for (int i = 0; i < 8; ++i) {
                float v = acc[i] + bias[base + i];
                t[i] = relu ? fmaxf(v, 0.0f) : v;
            }
            *(float4*)(out + base)     = make_float4(t[0], t[1], t[2], t[3]);
            *(float4*)(out + base + 4) = make_float4(t[4], t[5], t[6], t[7]);
        } else {                           // ragged final group only
#pragma unroll
            for (int i = 0; i < 8; ++i) {
                const int r = base + i;
                if (r < nrows) {
                    float v = acc[i] + bias[r];
                    out[r] = relu ? fmaxf(v, 0.0f) : v;
                }
            }
        }
    }
}

// ---- 1: embedding gather + input-vector assembly --------------------------
__global__ void k_init(const int* __restrict__ token, const float* __restrict__ hidden,
                       const float* __restrict__ embedding, float* __restrict__ ws) {
    const int t = token[0];
    const float* erow = embedding + (size_t)t * HID;
    for (int i = threadIdx.x; i < HID; i += blockDim.x) {
        const float e = erow[i];
        ws[WS_XIN + i]        = e;
        ws[WS_XCAT + i]       = e;
        ws[WS_XIN + HID + i]  = hidden[i];
    }
}

// ---- 2: attention logits (WMMA) + softmax + attn_applied ------------------
__global__ void k_attn(const float* __restrict__ attn_w, const float* __restrict__ attn_b,
                       const float* __restrict__ enc, float* __restrict__ ws,
                       float* __restrict__ d_out) {
    __shared__ float xs[2 * HID];
    __shared__ __align__(16) float alog[112];
    __shared__ float aw[128];
    __shared__ float red[2];
    const float* xin = ws + WS_XIN;
    for (int i = threadIdx.x; i < 2 * HID; i += blockDim.x) xs[i] = xin[i];
    __syncthreads();

    const int wave = threadIdx.x >> 5;     // 8 waves; 7 row-groups cover L=100
    if (wave < 7)
        gemv16_wave<false>(attn_w, xs, attn_b, alog, wave * 16, LEN, 2 * HID, 0);
    __syncthreads();

    if (threadIdx.x == 0) {                // L=100: serial reduce is negligible
        float mx = alog[0];
        for (int l = 1; l < LEN; ++l) mx = fmaxf(mx, alog[l]);
        float s = 0.0f;
        for (int l = 0; l < LEN; ++l) s += __expf(alog[l] - mx);
        red[0] = mx; red[1] = s;
    }
    __syncthreads();
    if (threadIdx.x < LEN) {
        const float w = __expf(alog[threadIdx.x] - red[0]) / red[1];
        aw[threadIdx.x] = w;
        d_out[VOC + HID + threadIdx.x] = w;        // attn_weights output
    }
    __syncthreads();

    for (int h = threadIdx.x; h < HID; h += blockDim.x) {   // coalesced over h
        float s = 0.0f;
#pragma unroll 4
        for (int l = 0; l < LEN; ++l) s += aw[l] * enc[l * HID + h];
        ws[WS_XCAT + HID + h] = s;
    }
}

// ---- 3: combine + ReLU (WMMA) ---------------------------------------------
__global__ void k_comb(const float* __restrict__ comb_w, const float* __restrict__ comb_b,
                       float* __restrict__ ws) {
    __shared__ float xs[2 * HID];
    const float* xcat = ws + WS_XCAT;
    for (int i = threadIdx.x; i < 2 * HID; i += blockDim.x) xs[i] = xcat[i];
    __syncthreads();
    const int g = blockIdx.x * (blockDim.x >> 5) + (threadIdx.x >> 5);
    if (g < HID / 16)
        gemv16_wave<false>(comb_w, xs, comb_b, ws + WS_X, g * 16, HID, 2 * HID, 1);
}

// ---- 4: GRU gate GEMVs (WMMA) ---------------------------------------------
__global__ void k_gru(const float* __restrict__ w_ih, const float* __restrict__ w_hh,
                      const float* __restrict__ b_ih, const float* __restrict__ b_hh,
                      const float* __restrict__ hidden, float* __restrict__ ws) {
    __shared__ float xs[2 * HID];          // [x | h0]
    const float* x = ws + WS_X;
    for (int i = threadIdx.x; i < HID; i += blockDim.x) {
        xs[i]       = x[i];
        xs[HID + i] = hidden[i];
    }
    __syncthreads();
    const int g = blockIdx.x * (blockDim.x >> 5) + (threadIdx.x >> 5);   // 0..383
    if (g < 192)
        gemv16_wave<false>(w_ih, xs, b_ih, ws + WS_GI, g * 16, 3 * HID, HID, 0);
    else if (g < 384)
        gemv16_wave<false>(w_hh, xs + HID, b_hh, ws + WS_GH, (g - 192) * 16, 3 * HID, HID, 0);
}

// ---- 5: GRU gate combine → h_new ------------------------------------------
__global__ void k_gate(const float* __restrict__ hidden, float* __restrict__ ws,
                       float* __restrict__ d_out) {
    const float* gi = ws + WS_GI;
    const float* gh = ws + WS_GH;
    for (int h = threadIdx.x; h < HID; h += blockDim.x) {
        const float r = 1.0f / (1.0f + __expf(-(gi[h] + gh[h])));
        const float z = 1.0f / (1.0f + __expf(-(gi[h + HID] + gh[h + HID])));
        const float n = tanhf(gi[h + 2 * HID] + r * gh[h + 2 * HID]);
        const float hn = (1.0f - z) * n + z * hidden[h];
        ws[WS_HNEW + h] = hn;
        d_out[VOC + h]  = hn;              // hidden output
    }
}

// ---- 6: output projection (WMMA, 206 MB stream — the roofline kernel) -----
// Rows < NT_SPLIT load RT (L2-resident across replays); rows >= NT_SPLIT load
// non-temporal so the streamed tail does not evict the resident portion.
__global__ void k_outproj(const float* __restrict__ out_w, const float* __restrict__ out_b,
                          const float* __restrict__ ws, float* __restrict__ logits) {
    __shared__ float xs[HID];
    const float* hnew = ws + WS_HNEW;
    for (int i = threadIdx.x; i < HID; i += blockDim.x) xs[i] = hnew[i];
    __syncthreads();
    const int g = blockIdx.x * (blockDim.x >> 5) + (threadIdx.x >> 5);
    const int ngroups = (VOC + 15) / 16;   // 3142
    if (g < ngroups) {
        const int row0 = g * 16;
        if (row0 < NT_SPLIT)
            gemv16_wave<false>(out_w, xs, out_b, logits, row0, VOC, HID, 0);
        else
            gemv16_wave<true>(out_w, xs, out_b, logits, row0, VOC, HID, 0);
    }
}

// ---- 7: log-sum-exp over V (201 KB, trivial vs weight stream) -------------
__global__ void k_logz(const float* __restrict__ logits, float* __restrict__ ws) {
    __shared__ float red[256];
    float mx = -3.4e38f;
    for (int v = threadIdx.x; v < VOC; v += 256) mx = fmaxf(mx, logits[v]);
    red[threadIdx.x] = mx;
    __syncthreads();
    for (int s = 128; s > 0; s >>= 1) {
        if (threadIdx.x < s) red[threadIdx.x] = fmaxf(red[threadIdx.x], red[threadIdx.x + s]);
        __syncthreads();
    }
    mx = red[0];
    __syncthreads();
    float sum = 0.0f;
    for (int v = threadIdx.x; v < VOC; v += 256) sum += __expf(logits[v] - mx);
    red[threadIdx.x] = sum;
    __syncthreads();
    for (int s = 128; s > 0; s >>= 1) {
        if (threadIdx.x < s) red[threadIdx.x] += red[threadIdx.x + s];
        __syncthreads();
    }
    if (threadIdx.x == 0) ws[WS_LOGZ] = mx + __logf(red[0]);
}

// ---- 8: log_softmax finalize (in place on d_out) --------------------------
__global__ void k_lsm(float* __restrict__ out, const float* __restrict__ ws) {
    const float lz = ws[WS_LOGZ];
    for (int v = blockIdx.x * blockDim.x + threadIdx.x; v < VOC; v += gridDim.x * blockDim.x)
        out[v] -= lz;
}

extern "C" void kernel_launch(void* const* d_in, const int* in_sizes, int n_in,
                              void* d_out, int out_size, void* d_ws, size_t ws_size,
                              hipStream_t stream) {
    const int*   token     = (const int*)  d_in[0];
    const float* hidden    = (const float*)d_in[1];
    const float* enc       = (const float*)d_in[2];
    const float* embedding = (const float*)d_in[3];
    const float* attn_w    = (const float*)d_in[4];
    const float* attn_b    = (const float*)d_in[5];
    const float* comb_w    = (const float*)d_in[6];
    const float* comb_b    = (const float*)d_in[7];
    const float* w_ih      = (const float*)d_in[8];
    const float* w_hh      = (const float*)d_in[9];
    const float* b_ih      = (const float*)d_in[10];
    const float* b_hh      = (const float*)d_in[11];
    const float* out_w     = (const float*)d_in[12];
    const float* out_b     = (const float*)d_in[13];
    float* out = (float*)d_out;
    float* ws  = (float*)d_ws;

    k_init   <<<1,   256, 0, stream>>>(token, hidden, embedding, ws);
    k_attn   <<<1,   256, 0, stream>>>(attn_w, attn_b, enc, ws, out);
    k_comb   <<<8,   256, 0, stream>>>(comb_w, comb_b, ws);
    k_gru    <<<48,  256, 0, stream>>>(w_ih, w_hh, b_ih, b_hh, hidden, ws);
    k_gate   <<<1,   256, 0, stream>>>(hidden, ws, out);
    k_outproj<<<393, 256, 0, stream>>>(out_w, out_b, ws, out);   // logits -> d_out[0:V]
    k_logz   <<<1,   256, 0, stream>>>(out, ws);
    k_lsm    <<<256, 256, 0, stream>>>(out, ws);
}